// tgate_hybrid_55679956025631
// MI455X (gfx1250) — compile-verified
//
#include <hip/hip_runtime.h>
#include <math.h>

typedef __attribute__((ext_vector_type(2))) float v2f;
typedef __attribute__((ext_vector_type(8))) float v8f;

// Problem shape (fixed by the reference)
constexpr int D      = 2048;     // reduction dim
constexpr int T      = 10;       // num types
constexpr int TP     = 16;       // T padded to WMMA N
constexpr int NROWS  = 4 * 4096; // B*S
// Tiling
constexpr int KC       = 256;        // K-chunk staged in LDS
constexpr int NCHUNK   = D / KC;     // 8 chunks
constexpr int WAVES    = 2;          // waves per block
constexpr int BLOCK    = WAVES * 32; // 64 threads
constexpr int ROWS_PB  = WAVES * 16; // 32 rows per block

__global__ __launch_bounds__(BLOCK)
void tgate_hybrid_fused_kernel(const float* __restrict__ x,
                               const float* __restrict__ Wc, const float* __restrict__ bc,
                               const float* __restrict__ Ws, const float* __restrict__ bs,
                               const float* __restrict__ Wg, const float* __restrict__ bg,
                               const float* __restrict__ alpha,
                               float* __restrict__ out)
{
    // B-fragment layout in LDS: lds_w[m][k/2][n][k&1] == W_m[k][n] (zero-padded n>=10).
    // Lane's B frag for WMMA step k0 is then one contiguous 8-byte ds_load.
    __shared__ float lds_w[3][KC / 2][TP][2];          // 48 KB
    __shared__ float lds_acc[ROWS_PB][3][TP];          // 6 KB (epilogue transpose)

    const int tid  = threadIdx.x;
    const int wid  = tid >> 5;
    const int lane = tid & 31;
    const int half = lane >> 4;   // 0: K=k0,k0+1 | 1: K=k0+2,k0+3
    const int l16  = lane & 15;

    const int rowBase = blockIdx.x * ROWS_PB + wid * 16;
    const float* px   = x + (size_t)(rowBase + l16) * D;

    const float* Wmat[3] = { Wc, Ws, Wg };

    v8f acc[3];
    #pragma unroll
    for (int m = 0; m < 3; ++m)
        #pragma unroll
        for (int r = 0; r < 8; ++r)
            acc[m][r] = 0.0f;

    for (int c = 0; c < NCHUNK; ++c) {
        const int k0 = c * KC;
        __syncthreads();   // previous chunk's LDS reads done before overwrite

        // Stage (and swizzle + pad) all three weight chunks into LDS.
        for (int idx = tid; idx < 3 * KC * TP; idx += BLOCK) {
            const int m = idx / (KC * TP);
            const int r = idx % (KC * TP);
            const int k = r / TP;
            const int n = r % TP;
            const float v = (n < T) ? Wmat[m][(size_t)(k0 + k) * T + n] : 0.0f;
            lds_w[m][k >> 1][n][k & 1] = v;
        }
        __syncthreads();

        // Prefetch next K-chunk of x (streamed once -> NT loads below).
        if (c + 1 < NCHUNK)
            __builtin_prefetch(px + k0 + KC + 2 * half, 0, 1);

        #pragma unroll 8
        for (int kk = 0; kk < KC; kk += 4) {
            // A fragment (16x4 f32): lanes 0-15 hold k,k+1; lanes 16-31 hold k+2,k+3.
            const v2f a = __builtin_nontemporal_load(
                (const v2f*)(px + k0 + kk + 2 * half));
            #pragma unroll
            for (int m = 0; m < 3; ++m) {
                // B fragment (4x16 f32): mirrored half-wave layout, one b64 LDS load.
                const v2f b = *(const v2f*)&lds_w[m][(kk >> 1) + half][l16][0];
                acc[m] = __builtin_amdgcn_wmma_f32_16x16x4_f32(
                    false, a, false, b, (short)0, acc[m], false, false);
            }
        }
    }

    // --- Epilogue: remap C layout (VGPR r, lane) -> row-per-thread via LDS ---
    __syncthreads();
    #pragma unroll
    for (int m = 0; m < 3; ++m)
        #pragma unroll
        for (int r = 0; r < 8; ++r)
            lds_acc[wid * 16 + r + 8 * half][m][l16] = acc[m][r];
    __syncthreads();

    if (tid < ROWS_PB) {
        const int row = tid;
        float Lc[T], Ls[T], Lg[T];
        #pragma unroll
        for (int n = 0; n < T; ++n) {
            Lc[n] = lds_acc[row][0][n] + bc[n];
            Ls[n] = lds_acc[row][1][n] + bs[n];
            Lg[n] = lds_acc[row][2][n] + bg[n];
        }
        // softmax over classifier logits
        float mx = Lc[0];
        #pragma unroll
        for (int n = 1; n < T; ++n) mx = fmaxf(mx, Lc[n]);
        float e[T], ssum = 0.0f;
        #pragma unroll
        for (int n = 0; n < T; ++n) { e[n] = __expf(Lc[n] - mx); ssum += e[n]; }
        // sigmoid gates
        float g[T];
        #pragma unroll
        for (int n = 0; n < T; ++n) g[n] = 1.0f / (1.0f + __expf(-Lg[n]));
        // dense dot: sum_n g[n] * soft[n]
        float softdot = 0.0f;
        #pragma unroll
        for (int n = 0; n < T; ++n) softdot += g[n] * e[n];
        softdot /= ssum;
        // top-2 of sparse logits (first-occurrence tie-break like lax.top_k)
        float v0 = -INFINITY, v1 = -INFINITY;
        int   i0 = 0, i1 = 0;
        #pragma unroll
        for (int n = 0; n < T; ++n) {
            const float t = Ls[n];
            if (t > v0)      { v1 = v0; i1 = i0; v0 = t; i0 = n; }
            else if (t > v1) { v1 = t;  i1 = n; }
        }
        const float s0 = 1.0f / (1.0f + __expf(v1 - v0));
        const float s1 = 1.0f - s0;
        const float sparsedot = g[i0] * s0 + g[i1] * s1;

        const float a = 1.0f / (1.0f + __expf(-alpha[0]));
        out[blockIdx.x * ROWS_PB + row] = a * sparsedot + (1.0f - a) * softdot;
    }
}

extern "C" void kernel_launch(void* const* d_in, const int* in_sizes, int n_in,
                              void* d_out, int out_size, void* d_ws, size_t ws_size,
                              hipStream_t stream) {
    (void)in_sizes; (void)n_in; (void)out_size; (void)d_ws; (void)ws_size;
    const float* x     = (const float*)d_in[0];
    const float* W_cls = (const float*)d_in[1];
    const float* b_cls = (const float*)d_in[2];
    const float* W_sp  = (const float*)d_in[3];
    const float* b_sp  = (const float*)d_in[4];
    const float* W_g   = (const float*)d_in[5];
    const float* b_g   = (const float*)d_in[6];
    const float* alpha = (const float*)d_in[7];
    float* out = (float*)d_out;

    dim3 grid(NROWS / ROWS_PB);   // 512 blocks
    dim3 block(BLOCK);            // 64 threads = 2 waves
    tgate_hybrid_fused_kernel<<<grid, block, 0, stream>>>(
        x, W_cls, b_cls, W_sp, b_sp, W_g, b_g, alpha, out);
}